// FastRPModel_27470610826033
// MI455X (gfx1250) — compile-verified
//
#include <hip/hip_runtime.h>
#include <hip/hip_bf16.h>
#include <math.h>

typedef __attribute__((ext_vector_type(2))) float v2f;
typedef __attribute__((ext_vector_type(4))) float v4f;
typedef __attribute__((ext_vector_type(8))) float v8f;

#define DIM 128

// ---------------- degree / normalization ----------------
__global__ void deg_kernel(const int* __restrict__ er, const float* __restrict__ ev,
                           float* __restrict__ deg, int nnz) {
    int e = blockIdx.x * blockDim.x + threadIdx.x;
    if (e < nnz) atomicAdd(&deg[er[e]], ev[e]);
}

__global__ void invdeg_kernel(float* __restrict__ deg, int n) {
    int i = blockIdx.x * blockDim.x + threadIdx.x;
    if (i < n) {
        float d = deg[i];
        deg[i] = (d > 0.0f) ? (1.0f / sqrtf(d)) : 0.0f;   // D^beta, beta = -0.5
    }
}

__global__ void count_kernel(const int* __restrict__ er, int* __restrict__ counts, int nnz) {
    int e = blockIdx.x * blockDim.x + threadIdx.x;
    if (e < nnz) atomicAdd(&counts[er[e]], 1);
}

// ---------------- single-block exclusive scan (rowptr) ----------------
__global__ __launch_bounds__(1024)
void scan_kernel(const int* __restrict__ counts, int* __restrict__ rowptr, int n) {
    __shared__ int sm[1024];
    __shared__ int carry;
    int tid = threadIdx.x;
    if (tid == 0) carry = 0;
    __syncthreads();
    for (int base = 0; base < n; base += 1024) {
        int v = (base + tid < n) ? counts[base + tid] : 0;
        sm[tid] = v;
        __syncthreads();
        for (int off = 1; off < 1024; off <<= 1) {
            int t = (tid >= off) ? sm[tid - off] : 0;
            __syncthreads();
            sm[tid] += t;
            __syncthreads();
        }
        int excl = sm[tid] - v;
        if (base + tid < n) rowptr[base + tid] = carry + excl;
        __syncthreads();
        if (tid == 0) carry += sm[1023];
        __syncthreads();
    }
    if (tid == 0) rowptr[n] = carry;
}

__global__ void scatter_kernel(const int* __restrict__ er, const int* __restrict__ ec,
                               const float* __restrict__ ev, int* __restrict__ cursor,
                               int* __restrict__ colS, float* __restrict__ valS, int nnz) {
    int e = blockIdx.x * blockDim.x + threadIdx.x;
    if (e >= nnz) return;
    int r = er[e];
    int pos = atomicAdd(&cursor[r], 1);
    colS[pos] = ec[e];
    valS[pos] = ev[e];
}

// ---------------- fused SpMM + L2-normalize + weighted emb accumulate ----------------
// One wave32 per row; each lane owns 4 of the 128 dims.
__global__ __launch_bounds__(256)
void spmm_kernel(const float* __restrict__ x, float* __restrict__ u,
                 float* __restrict__ emb, const float* __restrict__ invdeg,
                 const int* __restrict__ rowptr, const int* __restrict__ cols,
                 const float* __restrict__ vals, const float* __restrict__ fw,
                 int fidx, int F, int n) {
    int gid  = blockIdx.x * blockDim.x + threadIdx.x;
    int row  = gid >> 5;
    int lane = gid & 31;
    if (row >= n) return;

    int beg = rowptr[row], end = rowptr[row + 1];
    v4f acc = {0.f, 0.f, 0.f, 0.f};
    for (int e = beg; e < end; ++e) {
        int   c = cols[e];
        float v = vals[e];
        v4f xr = *reinterpret_cast<const v4f*>(x + (size_t)c * DIM + lane * 4);
        acc += v * xr;
    }
    acc *= invdeg[row];

    float ss = acc.x * acc.x + acc.y * acc.y + acc.z * acc.z + acc.w * acc.w;
    for (int off = 16; off > 0; off >>= 1) ss += __shfl_xor(ss, off, 32);
    float nrm  = sqrtf(ss);
    float inv  = 1.0f / fmaxf(nrm, 1e-12f);
    v4f ur = acc * inv;
    *reinterpret_cast<v4f*>(u + (size_t)row * DIM + lane * 4) = ur;

    // softmax(feature_weights)[fidx]
    float m = -1e30f;
    for (int k = 0; k < F; ++k) m = fmaxf(m, fw[k]);
    float s = 0.f, wf = 0.f;
    for (int k = 0; k < F; ++k) {
        float ek = expf(fw[k] - m);
        s += ek;
        if (k == fidx) wf = ek;
    }
    wf /= s;

    float* ep = emb + (size_t)row * DIM + lane * 4;
    v4f er = *reinterpret_cast<v4f*>(ep);
    er += wf * ur;
    *reinterpret_cast<v4f*>(ep) = er;
}

// ---------------- pair scoring via V_WMMA_F32_16X16X4_F32 ----------------
// One wave handles 16 pairs. Gram(D,D) diagonal = squared distance (exact f32).
// A(16x4) lane layout == B(4x16) lane layout for the symmetric case, so we pass
// the same registers for A and B.
__global__ __launch_bounds__(256)
void pairs_kernel(const float* __restrict__ emb,
                  const int* __restrict__ idx_i, const int* __restrict__ idx_j,
                  const float* __restrict__ intercept,
                  float* __restrict__ out, int npairs) {
    int wave = blockIdx.x * (blockDim.x >> 5) + (threadIdx.x >> 5);
    int lane = threadIdx.x & 31;
    int base = wave << 4;
    if (base >= npairs) return;          // wave-uniform: survivors keep full EXEC

    int m = lane & 15;
    int p = base + m;
    if (p >= npairs) p = npairs - 1;     // lane-level clamp (no divergence)

    const float* zi = emb + (size_t)idx_i[p] * DIM;
    const float* zj = emb + (size_t)idx_j[p] * DIM;
    int koff = (lane & 16) ? 2 : 0;      // lanes 0-15: K even pair {0,1}; 16-31: {2,3}

    v8f c = {0.f, 0.f, 0.f, 0.f, 0.f, 0.f, 0.f, 0.f};
#pragma unroll
    for (int k0 = 0; k0 < DIM; k0 += 4) {
        v2f a;
        a.x = zi[k0 + koff]     - zj[k0 + koff];
        a.y = zi[k0 + koff + 1] - zj[k0 + koff + 1];
        c = __builtin_amdgcn_wmma_f32_16x16x4_f32(
                /*neg_a=*/false, a, /*neg_b=*/false, a,
                /*c_mod=*/(short)0, c, /*reuse_a=*/false, /*reuse_b=*/false);
    }

    // diagonal: M=N=r. lanes 0-7 hold (M=r,N=lane) in c[r]; lanes 16-23 hold M=8+r.
    int r = -1, outp = -1;
    if (lane < 8)                       { r = lane;      outp = base + lane; }
    else if (lane >= 16 && lane < 24)   { r = lane - 16; outp = base + (lane - 8); }
    if (r >= 0 && outp < npairs) {
        float dist = c[0];
        dist = (r == 1) ? c[1] : dist;
        dist = (r == 2) ? c[2] : dist;
        dist = (r == 3) ? c[3] : dist;
        dist = (r == 4) ? c[4] : dist;
        dist = (r == 5) ? c[5] : dist;
        dist = (r == 6) ? c[6] : dist;
        dist = (r == 7) ? c[7] : dist;
        float z = intercept[0] - dist;
        out[outp] = 1.0f / (1.0f + expf(-z));
    }
}

// ---------------- launch ----------------
extern "C" void kernel_launch(void* const* d_in, const int* in_sizes, int n_in,
                              void* d_out, int out_size, void* d_ws, size_t ws_size,
                              hipStream_t stream) {
    (void)n_in; (void)ws_size; (void)out_size;
    const int*   edge_row  = (const int*)  d_in[0];
    const int*   edge_col  = (const int*)  d_in[1];
    const float* edge_val  = (const float*)d_in[2];
    const float* R_prime   = (const float*)d_in[3];
    const float* fw        = (const float*)d_in[4];
    const float* intercept = (const float*)d_in[5];
    const int*   idx_i     = (const int*)  d_in[6];
    const int*   idx_j     = (const int*)  d_in[7];
    float*       out       = (float*)d_out;

    const int nnz    = in_sizes[0];
    const int F      = in_sizes[4];
    const int n      = in_sizes[3] / DIM;
    const int npairs = in_sizes[6];

    // carve workspace (256-B aligned slices)
    char* w = (char*)d_ws;
    auto carve = [&](size_t bytes) -> char* {
        char* p = w;
        w += (bytes + 255) & ~(size_t)255;
        return p;
    };
    float* deg    = (float*)carve((size_t)n * 4);           // becomes inv_deg in place
    int*   counts = (int*)  carve((size_t)n * 4);
    int*   rowptr = (int*)  carve((size_t)(n + 1) * 4);
    int*   cursor = (int*)  carve((size_t)n * 4);
    int*   colS   = (int*)  carve((size_t)nnz * 4);
    float* valS   = (float*)carve((size_t)nnz * 4);
    float* u_a    = (float*)carve((size_t)n * DIM * 4);
    float* u_b    = (float*)carve((size_t)n * DIM * 4);
    float* emb    = (float*)carve((size_t)n * DIM * 4);

    hipMemsetAsync(deg,    0, (size_t)n * 4,       stream);
    hipMemsetAsync(counts, 0, (size_t)n * 4,       stream);
    hipMemsetAsync(emb,    0, (size_t)n * DIM * 4, stream);

    const int tb = 256;
    const int gb_e = (nnz + tb - 1) / tb;
    deg_kernel   <<<gb_e, tb, 0, stream>>>(edge_row, edge_val, deg, nnz);
    count_kernel <<<gb_e, tb, 0, stream>>>(edge_row, counts, nnz);
    invdeg_kernel<<<(n + tb - 1) / tb, tb, 0, stream>>>(deg, n);
    scan_kernel  <<<1, 1024, 0, stream>>>(counts, rowptr, n);
    hipMemcpyAsync(cursor, rowptr, (size_t)n * 4, hipMemcpyDeviceToDevice, stream);
    scatter_kernel<<<gb_e, tb, 0, stream>>>(edge_row, edge_col, edge_val,
                                            cursor, colS, valS, nnz);

    const float* xin = R_prime;
    float* bufs[2] = {u_a, u_b};
    const int gb_r = (n * 32 + tb - 1) / tb;   // one wave per row
    for (int f = 0; f < F; ++f) {
        float* uo = bufs[f & 1];
        spmm_kernel<<<gb_r, tb, 0, stream>>>(xin, uo, emb, deg, rowptr, colS, valS,
                                             fw, f, F, n);
        xin = uo;
    }

    const int waves = (npairs + 15) / 16;
    const int gb_p  = (waves * 32 + tb - 1) / tb;
    pairs_kernel<<<gb_p, tb, 0, stream>>>(emb, idx_i, idx_j, intercept, out, npairs);
}